// MambaVisionMixer_29549374997019
// MI455X (gfx1250) — compile-verified
//
#include <hip/hip_runtime.h>

// ---- problem dims (fixed by reference) ----
#define LSEQ   2048
#define BATCHN 2
#define DMODEL 768
#define DHALF  768
#define DINNER 1536
#define DSTATE 16
#define DTRANK 48
#define XDBLW  80            // DT_RANK + 2*D_STATE
#define MROWS  (BATCHN*LSEQ) // 4096 fused (b,l) rows

typedef __attribute__((ext_vector_type(2))) float v2f;
typedef __attribute__((ext_vector_type(8))) float v8f;

static __device__ __forceinline__ v8f wmma4(v2f a, v2f b, v8f c) {
  return __builtin_amdgcn_wmma_f32_16x16x4_f32(false, a, false, b, (short)0, c,
                                               false, false);
}

// ---------------------------------------------------------------------------
// 16x16-per-wave GEMM (used only for the tiny x_proj GEMM, N=80).
// Out[m][n] = sum_k A[m][k] * W[n][k]
// ---------------------------------------------------------------------------
__global__ __launch_bounds__(256) void wmma_gemm_xWT_16(
    const float* __restrict__ A, const float* __restrict__ W,
    float* __restrict__ Out, int M, int N, int K, int lda, int ldw, int ldo)
{
  const int wave = threadIdx.x >> 5;
  const int lane = threadIdx.x & 31;
  const int l16  = lane & 15;
  const int half = lane >> 4;
  const int ntn  = N >> 4;
  const int wid  = blockIdx.x * (blockDim.x >> 5) + wave;
  if (wid >= (M >> 4) * ntn) return;   // wave-uniform: EXEC all-1 for WMMA
  const int m0 = (wid / ntn) << 4;
  const int n0 = (wid % ntn) << 4;

  const float2* arow = reinterpret_cast<const float2*>(A + (size_t)(m0 + l16) * lda);
  const float2* brow = reinterpret_cast<const float2*>(W + (size_t)(n0 + l16) * ldw);

  v8f acc = {};
  for (int k0 = 0; k0 < K; k0 += 16) {
#pragma unroll
    for (int i = 0; i < 4; ++i) {
      const int kh = (k0 >> 1) + 2 * i + half;
      float2 af = arow[kh];
      float2 bf = brow[kh];
      v2f av = { af.x, af.y };
      v2f bv = { bf.x, bf.y };
      acc = wmma4(av, bv, acc);
    }
  }
  float* crow = Out + (size_t)(m0 + 8 * half) * ldo + (n0 + l16);
#pragma unroll
  for (int j = 0; j < 8; ++j) crow[(size_t)j * ldo] = acc[j];
}

// ---------------------------------------------------------------------------
// 32x32-per-wave register-blocked GEMM: 4 accumulators, A/B fragments reused
// across 2 tiles each -> 4 WMMAs per 4 float2 loads (2x less L2 traffic/FLOP).
// Requires M%32==0, N%32==0, K%16==0, even leading dims.
// ---------------------------------------------------------------------------
__global__ __launch_bounds__(256) void wmma_gemm_xWT_32(
    const float* __restrict__ A, const float* __restrict__ W,
    float* __restrict__ Out, const float* __restrict__ bias, float bias_scale,
    int M, int N, int K, int lda, int ldw, int ldo)
{
  const int wave = threadIdx.x >> 5;
  const int lane = threadIdx.x & 31;
  const int l16  = lane & 15;
  const int half = lane >> 4;
  const int ntn  = N >> 5;
  const int wid  = blockIdx.x * (blockDim.x >> 5) + wave;
  if (wid >= (M >> 5) * ntn) return;   // wave-uniform
  const int m0 = (wid / ntn) << 5;
  const int n0 = (wid % ntn) << 5;

  const float2* arow0 = reinterpret_cast<const float2*>(A + (size_t)(m0 + l16) * lda);
  const float2* arow1 = reinterpret_cast<const float2*>(A + (size_t)(m0 + 16 + l16) * lda);
  const float2* brow0 = reinterpret_cast<const float2*>(W + (size_t)(n0 + l16) * ldw);
  const float2* brow1 = reinterpret_cast<const float2*>(W + (size_t)(n0 + 16 + l16) * ldw);

  v8f acc00 = {}, acc01 = {}, acc10 = {}, acc11 = {};
  for (int k0 = 0; k0 < K; k0 += 16) {
    if (k0 + 64 < K) {                       // prefetch 64 k (4 iterations) ahead
      const int ph = ((k0 + 64) >> 1) + half;
      __builtin_prefetch((const void*)(arow0 + ph), 0, 1);
      __builtin_prefetch((const void*)(arow1 + ph), 0, 1);
      __builtin_prefetch((const void*)(brow0 + ph), 0, 1);
      __builtin_prefetch((const void*)(brow1 + ph), 0, 1);
    }
#pragma unroll
    for (int i = 0; i < 4; ++i) {
      const int kh = (k0 >> 1) + 2 * i + half;   // (k0 + 4i + 2*half)/2
      float2 a0 = arow0[kh], a1 = arow1[kh];
      float2 b0 = brow0[kh], b1 = brow1[kh];
      v2f av0 = { a0.x, a0.y }, av1 = { a1.x, a1.y };
      v2f bv0 = { b0.x, b0.y }, bv1 = { b1.x, b1.y };
      acc00 = wmma4(av0, bv0, acc00);
      acc01 = wmma4(av0, bv1, acc01);
      acc10 = wmma4(av1, bv0, acc10);
      acc11 = wmma4(av1, bv1, acc11);
    }
  }

  const float badd0 = bias ? bias_scale * bias[n0 + l16]      : 0.0f;
  const float badd1 = bias ? bias_scale * bias[n0 + 16 + l16] : 0.0f;
  float* c0 = Out + (size_t)(m0 + 8 * half) * ldo + (n0 + l16);
  float* c1 = Out + (size_t)(m0 + 16 + 8 * half) * ldo + (n0 + l16);
#pragma unroll
  for (int j = 0; j < 8; ++j) {
    c0[(size_t)j * ldo]      = acc00[j] + badd0;
    c0[(size_t)j * ldo + 16] = acc01[j] + badd1;
    c1[(size_t)j * ldo]      = acc10[j] + badd0;
    c1[(size_t)j * ldo + 16] = acc11[j] + badd1;
  }
}

// ---------------------------------------------------------------------------
// Depthwise conv (k=4, pad=(1,2)) + SiLU over xz (B,L,1536).
// x half -> xs (B,L,768); z half -> yfull[:, :, 768:1536].
// ---------------------------------------------------------------------------
__global__ __launch_bounds__(256) void conv_silu_kernel(
    const float* __restrict__ xz,
    const float* __restrict__ wx, const float* __restrict__ bx,
    const float* __restrict__ wz, const float* __restrict__ bz,
    float* __restrict__ xs, float* __restrict__ yfull)
{
  const int idx = blockIdx.x * blockDim.x + threadIdx.x;   // over B*L*1536
  const int c  = idx % DINNER;
  const int bl = idx / DINNER;
  const int l  = bl % LSEQ;
  const int b  = bl / LSEQ;

  const bool isx = (c < DHALF);
  const int  cc  = isx ? c : c - DHALF;
  const float* w = (isx ? wx : wz) + cc * 4;
  float acc = isx ? bx[cc] : bz[cc];

#pragma unroll
  for (int j = 0; j < 4; ++j) {
    const int ll = l + j - 1;                    // 'same' padding: 1 left, 2 right
    float v = (ll >= 0 && ll < LSEQ)
                ? xz[((size_t)b * LSEQ + ll) * DINNER + c] : 0.0f;
    acc += w[j] * v;
  }
  const float s = acc / (1.0f + __expf(-acc));   // SiLU

  const size_t row = (size_t)b * LSEQ + l;
  if (isx) xs[row * DHALF + cc] = s;
  else     yfull[row * DINNER + DHALF + cc] = s;
}

// ---------------------------------------------------------------------------
// Selective scan, one LANE per (b, d, n) state: 16 lanes form one channel.
// h[n] recurrence is a scalar per lane; y = sum_n h[n]*C[n] via 4x shfl_xor
// (masks < 16 keep the reduction inside each 16-lane state group on wave32).
// dt (softplus pre-applied), u, B, C staged in LDS per 32-timestep chunk.
// ---------------------------------------------------------------------------
__global__ __launch_bounds__(256) void scan_kernel(
    const float* __restrict__ xs,   const float* __restrict__ dtv,
    const float* __restrict__ xdbl, const float* __restrict__ A_log,
    const float* __restrict__ Dp,   float* __restrict__ yfull)
{
  __shared__ float sDT[32][16];   // softplus(dt)
  __shared__ float sU [32][16];
  __shared__ float sB [32][16];
  __shared__ float sC [32][16];

  const int b  = blockIdx.y;
  const int d0 = blockIdx.x * 16;          // 16 channels per block
  const int ch = threadIdx.x >> 4;         // 0..15 channel-in-block
  const int n  = threadIdx.x & 15;         // state index
  const int d  = d0 + ch;

  const float Ad = -__expf(A_log[d * DSTATE + n]);
  const float Dv = Dp[d];
  float h = 0.0f;
  const size_t baseBL = (size_t)b * LSEQ;

  for (int t0 = 0; t0 < LSEQ; t0 += 32) {
    __syncthreads();
    for (int i = threadIdx.x; i < 512; i += 256) {
      const int tt = i >> 4, j = i & 15;
      const size_t row  = baseBL + t0 + tt;
      const size_t offd = row * DHALF + d0 + j;
      const float  dr   = dtv[offd];
      sDT[tt][j] = (dr > 20.0f) ? dr : __logf(1.0f + __expf(dr));
      sU [tt][j] = xs[offd];
      sB [tt][j] = xdbl[row * XDBLW + DTRANK + j];
      sC [tt][j] = xdbl[row * XDBLW + DTRANK + DSTATE + j];
    }
    __syncthreads();

    for (int tt = 0; tt < 32; ++tt) {
      const float delta = sDT[tt][ch];
      const float u     = sU[tt][ch];
      const float dA    = __expf(delta * Ad);
      h = dA * h + (delta * u) * sB[tt][n];
      float p = h * sC[tt][n];
      p += __shfl_xor(p, 1);
      p += __shfl_xor(p, 2);
      p += __shfl_xor(p, 4);
      p += __shfl_xor(p, 8);
      if (n == 0)
        yfull[(baseBL + t0 + tt) * DINNER + d] = p + u * Dv;
    }
  }
}

// ---------------------------------------------------------------------------
// Host orchestration.  Workspace (floats):
//   xz: 6291456 | xs: 3145728 | yfull: 6291456 | xdbl: 327680 | dtv: 3145728
// total ~76.8 MB.
// ---------------------------------------------------------------------------
extern "C" void kernel_launch(void* const* d_in, const int* in_sizes, int n_in,
                              void* d_out, int out_size, void* d_ws, size_t ws_size,
                              hipStream_t stream)
{
  const float* hidden    = (const float*)d_in[0];
  const float* in_proj_w = (const float*)d_in[1];
  const float* x_proj_w  = (const float*)d_in[2];
  const float* dt_proj_w = (const float*)d_in[3];
  const float* dt_proj_b = (const float*)d_in[4];
  const float* A_log     = (const float*)d_in[5];
  const float* D_param   = (const float*)d_in[6];
  const float* conv_x_w  = (const float*)d_in[7];
  const float* conv_x_b  = (const float*)d_in[8];
  const float* conv_z_w  = (const float*)d_in[9];
  const float* conv_z_b  = (const float*)d_in[10];
  const float* out_proj_w= (const float*)d_in[11];

  float* ws    = (float*)d_ws;
  float* xz    = ws;
  float* xs    = xz    + (size_t)MROWS * DINNER;
  float* yfull = xs    + (size_t)MROWS * DHALF;
  float* xdbl  = yfull + (size_t)MROWS * DINNER;
  float* dtv   = xdbl  + (size_t)MROWS * XDBLW;

  // 1) xz = hidden @ in_proj_w^T         (4096 x 1536, K=768)
  {
    const int tiles = (MROWS / 32) * (DINNER / 32);
    wmma_gemm_xWT_32<<<(tiles + 7) / 8, 256, 0, stream>>>(
        hidden, in_proj_w, xz, nullptr, 0.0f,
        MROWS, DINNER, DMODEL, DMODEL, DMODEL, DINNER);
  }

  // 2) depthwise conv + SiLU -> xs, and z -> yfull[:, 768:]
  conv_silu_kernel<<<(MROWS * DINNER) / 256, 256, 0, stream>>>(
      xz, conv_x_w, conv_x_b, conv_z_w, conv_z_b, xs, yfull);

  // 3) x_dbl = xs @ x_proj_w^T           (4096 x 80, K=768) -- N not %32
  {
    const int tiles = (MROWS / 16) * (XDBLW / 16);
    wmma_gemm_xWT_16<<<(tiles + 7) / 8, 256, 0, stream>>>(
        xs, x_proj_w, xdbl, MROWS, XDBLW, DMODEL, DMODEL, DMODEL, XDBLW);
  }

  // 4) dtv = x_dbl[:, :48] @ dt_proj_w^T + 2*dt_proj_b (reference adds bias twice)
  {
    const int tiles = (MROWS / 32) * (DHALF / 32);
    wmma_gemm_xWT_32<<<(tiles + 7) / 8, 256, 0, stream>>>(
        xdbl, dt_proj_w, dtv, dt_proj_b, 2.0f,
        MROWS, DHALF, DTRANK, XDBLW, DTRANK, DHALF);
  }

  // 5) selective scan -> yfull[:, :768]   (one lane per (b,d,n) state)
  scan_kernel<<<dim3(DHALF / 16, BATCHN), 256, 0, stream>>>(
      xs, dtv, xdbl, A_log, D_param, yfull);

  // 6) out = yfull @ out_proj_w^T        (4096 x 768, K=1536)
  {
    const int tiles = (MROWS / 32) * (DMODEL / 32);
    wmma_gemm_xWT_32<<<(tiles + 7) / 8, 256, 0, stream>>>(
        yfull, out_proj_w, (float*)d_out, nullptr, 0.0f,
        MROWS, DMODEL, DINNER, DINNER, DINNER, DMODEL);
  }
}